// Detect_37280316129505
// MI455X (gfx1250) — compile-verified
//
#include <hip/hip_runtime.h>
#include <stdint.h>

// ---- problem constants (match reference) ----
#define B_      4
#define P_      32768
#define TOPK    5000
#define NW      79            // u64 words per suppression row (79*64 = 5056 bits)
#define TKP     5056          // padded top-K (NW*64)
#define IT_     313           // i-tiles of 16 rows (313*16 = 5008)
#define ROWS_PAD 5008
#define NMS_T   0.3f
#define CONF_T  0.05f
#define VAR0_   0.1f
#define VAR1_   0.2f

typedef float v2f __attribute__((ext_vector_type(2)));
typedef float v8f __attribute__((ext_vector_type(8)));

__device__ __forceinline__ uint32_t f32_ordkey(float f) {
    uint32_t u = __float_as_uint(f);
    return (u & 0x80000000u) ? ~u : (u | 0x80000000u);
}

__device__ __forceinline__ uint64_t shfl64(uint64_t v, int src) {
    int lo = __shfl((int)(uint32_t)(v & 0xffffffffull), src, 32);
    int hi = __shfl((int)(uint32_t)(v >> 32), src, 32);
    return ((uint64_t)(uint32_t)hi << 32) | (uint32_t)lo;
}

// ---------------- Stage 1: decode boxes + build sortable keys ----------------
__global__ void decode_kernel(const float* __restrict__ loc,
                              const float* __restrict__ conf,
                              const float* __restrict__ prior,
                              float4* __restrict__ boxes,
                              uint64_t* __restrict__ keys) {
    int t = blockIdx.x * blockDim.x + threadIdx.x;
    if (t >= B_ * P_) return;
    int p = t & (P_ - 1);
    float4 pr = ((const float4*)prior)[p];
    float4 lo = ((const float4*)loc)[t];
    float cx = pr.x + lo.x * VAR0_ * pr.z;
    float cy = pr.y + lo.y * VAR0_ * pr.w;
    float w  = pr.z * expf(lo.z * VAR1_);
    float h  = pr.w * expf(lo.w * VAR1_);
    boxes[t] = make_float4(cx - 0.5f * w, cy - 0.5f * h,
                           cx + 0.5f * w, cy + 0.5f * h);
    float s = conf[t * 2 + 1];
    float masked = (s > CONF_T) ? s : -__builtin_inff();
    // descending by score, ascending by index on ties (stable argsort semantics)
    keys[t] = ((uint64_t)f32_ordkey(masked) << 32) | (uint32_t)(~(uint32_t)p);
}

// ---------------- Stage 2: full in-LDS bitonic sort (256KB LDS!) + gather ----
__global__ void sort_topk_kernel(const uint64_t* __restrict__ keys,
                                 const float* __restrict__ conf,
                                 const float4* __restrict__ boxes,
                                 float4* __restrict__ topBoxes,
                                 float* __restrict__ topScores) {
    extern __shared__ uint64_t sk[];              // P_ u64 = 256 KB (CDNA5 320KB LDS)
    const int img = blockIdx.x;
    const int tid = threadIdx.x;
    const int nt  = blockDim.x;
    for (int i = tid; i < P_; i += nt) sk[i] = keys[img * P_ + i];
    __syncthreads();
    // bitonic sort, descending
    for (int k = 2; k <= P_; k <<= 1) {
        for (int j = k >> 1; j > 0; j >>= 1) {
            for (int i = tid; i < P_; i += nt) {
                int l = i ^ j;
                if (l > i) {
                    uint64_t a = sk[i], b = sk[l];
                    bool up = (i & k) == 0;
                    bool sw = up ? (a < b) : (a > b);
                    if (sw) { sk[i] = b; sk[l] = a; }
                }
            }
            __syncthreads();
        }
    }
    for (int kk = tid; kk < TKP; kk += nt) {
        float4 bx = make_float4(0.f, 0.f, 0.f, 0.f);
        float  s  = -1e30f;
        if (kk < TOPK) {
            uint32_t p = (~(uint32_t)sk[kk]) & (P_ - 1);
            s  = conf[(img * P_ + p) * 2 + 1];
            bx = boxes[img * P_ + p];
        }
        topScores[img * TKP + kk] = s;
        topBoxes [img * TKP + kk] = bx;
    }
}

// ---------------- Stage 3: pairwise IoU -> suppression bitmask (WMMA tiles) --
// Each wave handles a 16-row x 64-col tile. The (area_i + area_j) outer-sum for
// each 16x16 subtile is computed with one V_WMMA_F32_16X16X4_F32:
//   A(16x4) row m = [area_i[m], 1, 0, 0];  B(4x16) = [[1..1],[area_j],0,0]
// f32 layouts: A lanes0-15 hold {K0,K1}, lanes16-31 {K2,K3}=0; same for B.
__global__ void iou_mask_kernel(const float4* __restrict__ topBoxes,
                                uint64_t* __restrict__ supMask) {
    const int wg   = (blockIdx.x * blockDim.x + threadIdx.x) >> 5;
    const int lane = threadIdx.x & 31;
    const int total = B_ * IT_ * NW;
    if (wg >= total) return;                       // uniform per wave (EXEC stays full)
    const int img = wg / (IT_ * NW);
    const int rem = wg % (IT_ * NW);
    const int iT  = rem / NW;
    const int wT  = rem % NW;
    const int i0  = iT * 16;
    const int j0  = wT * 64;
    const float4* tb = topBoxes + (size_t)img * TKP;
    const int half = lane >> 4;
    const int ln   = lane & 15;

    float4 bi = tb[i0 + ln];
    float areaI = (bi.z - bi.x) * (bi.w - bi.y);
    v2f A;
    A.x = half ? 0.0f : areaI;                     // K0 (lanes 0-15) / K2 = 0
    A.y = half ? 0.0f : 1.0f;                      // K1 (lanes 0-15) / K3 = 0

    // hoist the 8 i-boxes this lane compares against (s-invariant)
    float4 bmv[8];
    #pragma unroll
    for (int v = 0; v < 8; ++v) bmv[v] = tb[i0 + (half << 3) + v];

    // lane -> row ownership for the packing phase (branch-free select chain)
    const int vn   = ln & 7;                       // which ballot word holds my row
    const int hsel = (ln >> 3) << 4;               // 0 or 16: low/high half of ballot

    uint64_t rowmask = 0;
    for (int s = 0; s < 4; ++s) {
        float4 bj = tb[j0 + s * 16 + ln];
        float areaJ = (bj.z - bj.x) * (bj.w - bj.y);
        v2f Bv;
        Bv.x = half ? 0.0f : 1.0f;                 // B row K0 = ones
        Bv.y = half ? 0.0f : areaJ;                // B row K1 = area_j
        v8f C = {};
        v8f D = __builtin_amdgcn_wmma_f32_16x16x4_f32(
            false, A, false, Bv, (short)0, C, false, false);
        // D element (m = v + 8*half, n = ln) = area_i[m] + area_j[n]
        // ballot the IoU predicate directly: bl[v] holds rows v (lo16) & v+8 (hi16)
        unsigned bl[8];
        #pragma unroll
        for (int v = 0; v < 8; ++v) {
            float4 bm = bmv[v];
            float iw = fmaxf(fminf(bm.z, bj.z) - fmaxf(bm.x, bj.x), 0.0f);
            float ih = fmaxf(fminf(bm.w, bj.w) - fmaxf(bm.y, bj.y), 0.0f);
            float inter = iw * ih;
            float uni = D[v] - inter;              // union via WMMA outer-sum
            bl[v] = (unsigned)__ballot(inter > NMS_T * uni);
        }
        // branch-free: pick my ballot word with a cndmask chain, then my 16 bits
        unsigned sel = bl[0];
        sel = (vn == 1) ? bl[1] : sel;
        sel = (vn == 2) ? bl[2] : sel;
        sel = (vn == 3) ? bl[3] : sel;
        sel = (vn == 4) ? bl[4] : sel;
        sel = (vn == 5) ? bl[5] : sel;
        sel = (vn == 6) ? bl[6] : sel;
        sel = (vn == 7) ? bl[7] : sel;
        uint64_t take = (uint64_t)((sel >> hsel) & 0xFFFFu);
        rowmask |= take << (s * 16);
    }
    if (half == 0)
        supMask[((uint64_t)img * ROWS_PAD + (i0 + ln)) * NW + wT] = rowmask;
}

// ---------------- Stage 4a: zero the whole output -----------------------------
__global__ void zero_kernel(float* __restrict__ out, int n) {
    int t = blockIdx.x * blockDim.x + threadIdx.x;
    if (t < n) out[t] = 0.0f;
}

// ---------------- Stage 4b: serial greedy scan (1 wave / image) + scatter -----
__global__ void nms_scan_kernel(const float* __restrict__ topScores,
                                const float4* __restrict__ topBoxes,
                                const uint64_t* __restrict__ supMask,
                                float* __restrict__ out) {
    const int img  = blockIdx.x;
    const int lane = threadIdx.x;                  // blockDim.x == 32
    __shared__ uint64_t keptLds[NW];
    uint64_t supw[3]  = {0, 0, 0};                 // lane owns words lane, lane+32, lane+64
    uint64_t keptw[3] = {0, 0, 0};
    const float*    sc = topScores + (size_t)img * TKP;
    const uint64_t* sm = supMask + (uint64_t)img * ROWS_PAD * NW;

    for (int j = 0; j < TOPK; ++j) {
        // prefetch an upcoming suppression row (global_prefetch_b8) so the
        // dependent OR loads on the serial chain hit hot cache lines
        if (j + 4 < TOPK) {
            const char* rn = (const char*)(sm + (uint64_t)(j + 4) * NW);
            __builtin_prefetch(rn + lane * 24, 0, 0);   // 32 lanes x 24B covers 632B row
        }
        int w = j >> 6, slot = w >> 5, owner = w & 31;   // all uniform
        uint64_t mine = (slot == 0) ? supw[0] : (slot == 1) ? supw[1] : supw[2];
        uint64_t sv = shfl64(mine, owner);
        bool suppressed = (sv >> (j & 63)) & 1ull;
        bool keptj = (sc[j] > CONF_T) && !suppressed;    // uniform branch
        if (keptj) {
            const uint64_t* row = sm + (uint64_t)j * NW;
            #pragma unroll
            for (int t = 0; t < 3; ++t) {
                int w2 = lane + t * 32;
                if (w2 < NW) supw[t] |= row[w2];
            }
            // branch-free on the lane dimension (slot is uniform)
            uint64_t setbit = (uint64_t)(lane == owner) << (j & 63);
            if (slot == 0)      keptw[0] |= setbit;
            else if (slot == 1) keptw[1] |= setbit;
            else                keptw[2] |= setbit;
        }
    }
    #pragma unroll
    for (int t = 0; t < 3; ++t) {
        int w2 = lane + t * 32;
        if (w2 < NW) keptLds[w2] = keptw[t];
    }
    __syncthreads();
    // compacted scatter: ballot-based prefix within each 32-chunk
    int base = 0;
    const float4* tb = topBoxes + (size_t)img * TKP;
    for (int jb = 0; jb < TOPK; jb += 32) {
        int j = jb + lane;
        bool k = (j < TOPK) && ((keptLds[j >> 6] >> (j & 63)) & 1ull);
        unsigned mask = (unsigned)__ballot(k);
        if (k) {
            int pos = base + __popc(mask & ((1u << lane) - 1u));
            float4 b = tb[j];
            float* o = out + ((size_t)(img * 2 + 1) * TOPK + pos) * 5;
            o[0] = sc[j]; o[1] = b.x; o[2] = b.y; o[3] = b.z; o[4] = b.w;
        }
        base += __popc(mask);
    }
}

// ---------------- host launch -------------------------------------------------
extern "C" void kernel_launch(void* const* d_in, const int* in_sizes, int n_in,
                              void* d_out, int out_size, void* d_ws, size_t ws_size,
                              hipStream_t stream) {
    (void)in_sizes; (void)n_in; (void)ws_size;
    const float* loc   = (const float*)d_in[0];
    const float* conf  = (const float*)d_in[1];
    const float* prior = (const float*)d_in[2];
    float* out = (float*)d_out;

    char* ws = (char*)d_ws;
    float4*   boxes     = (float4*)ws;                                   // B*P*16B  = 2 MB
    uint64_t* keys      = (uint64_t*)(ws + (size_t)B_ * P_ * 16);        // B*P*8B   = 1 MB
    char* p2            = ws + (size_t)B_ * P_ * 24;
    float4*   topBoxes  = (float4*)p2;                                   // B*TKP*16B
    float*    topScores = (float*)(p2 + (size_t)B_ * TKP * 16);          // B*TKP*4B
    uint64_t* supMask   = (uint64_t*)(p2 + (size_t)B_ * TKP * 20);       // B*5008*79*8B ~ 12.6 MB

    int n1 = B_ * P_;
    decode_kernel<<<(n1 + 255) / 256, 256, 0, stream>>>(loc, conf, prior, boxes, keys);

    // one workgroup per image; 256KB dynamic LDS (CDNA5 WGP has 320KB)
    sort_topk_kernel<<<B_, 1024, (size_t)P_ * sizeof(uint64_t), stream>>>(
        keys, conf, boxes, topBoxes, topScores);

    int totalWaves = B_ * IT_ * NW;
    int blocks3 = (totalWaves * 32 + 255) / 256;
    iou_mask_kernel<<<blocks3, 256, 0, stream>>>(topBoxes, supMask);

    zero_kernel<<<(out_size + 255) / 256, 256, 0, stream>>>(out, out_size);

    nms_scan_kernel<<<B_, 32, 0, stream>>>(topScores, topBoxes, supMask, out);
}